// PartitionedCliffordAlgebra_29454885716254
// MI455X (gfx1250) — compile-verified
//
#include <hip/hip_runtime.h>
#include <stdint.h>

#define DIM 64
#define NBITS 6
#define ROWS_PER_BLOCK 128
#define THREADS 128
#define LDS_STRIDE 68   // 64 data floats + 4 pad floats inserted by TDM (pad_interval=64DW, pad_amount=4DW)

// ---------------- compile-time Cayley sign table for Cl(3,2,1) ----------------
struct SignTab { signed char s[DIM][DIM]; };

constexpr int pcnt(int v) { int c = 0; while (v) { c += v & 1; v >>= 1; } return c; }

constexpr int sign_ab(int a, int b) {
    int swap = 0;
    for (int bit = 0; bit < NBITS; ++bit)
        if ((a >> bit) & 1) swap += pcnt(b & ((1 << bit) - 1));
    int s = (swap & 1) ? -1 : 1;
    if (pcnt(a & b & 0x18) & 1) s = -s;   // negative dims: bits 3,4 (Q=2)
    if (a & b & 0x20) s = 0;              // null dim: bit 5 (R=1)
    return s;
}

constexpr SignTab make_tab() {
    SignTab t{};
    for (int k = 0; k < DIM; ++k)
        for (int a = 0; a < DIM; ++a)
            t.s[k][a] = (signed char)sign_ab(a, a ^ k);   // SG[k][a] = sign(a, a^k)
    return t;
}
constexpr SignTab TAB = make_tab();

// ---------------- Tensor Data Mover: 2D tile (rows x 64 f32) global -> LDS ----------------
typedef unsigned int u32x4 __attribute__((ext_vector_type(4)));
typedef int          i32x4 __attribute__((ext_vector_type(4)));
typedef int          i32x8 __attribute__((ext_vector_type(8)));

__device__ static inline void tdm_load_tile(const float* gsrc, uint32_t lds_byte_off, int rows) {
    const uint64_t ga = (uint64_t)(uintptr_t)gsrc;
    u32x4 g0;
    g0[0] = 1u;                                                // count=1, user descriptor
    g0[1] = lds_byte_off;                                      // lds_addr (bytes)
    g0[2] = (uint32_t)ga;                                      // global_addr[31:0]
    g0[3] = (uint32_t)((ga >> 32) & 0x01FFFFFFu) | (2u << 30); // global_addr[56:32] | type=2
    i32x8 g1;
    // data_size=4B (code 2), pad_enable=1, pad_interval=64DW (code 5), pad_amount=4DW (code 3)
    g1[0] = (int)((2u << 16) | (1u << 20) | (5u << 22) | (3u << 25));
    g1[1] = (int)(64u << 16);                                  // tensor_dim0 = 64 (low 16 bits in [31:16])
    g1[2] = (int)((uint32_t)rows << 16);                       // tensor_dim1[15:0] in [31:16]
    g1[3] = (int)(64u << 16);                                  // tile_dim0 = 64 in [31:16]
    g1[4] = rows;                                              // tile_dim1 = rows, tile_dim2 = 0
    g1[5] = 64;                                                // tensor_dim0_stride = 64 elements
    g1[6] = 0;
    g1[7] = 0;
    i32x4 z4 = {0, 0, 0, 0};
#if defined(__clang_major__) && (__clang_major__ >= 23)
    i32x8 z8 = {0, 0, 0, 0, 0, 0, 0, 0};
    __builtin_amdgcn_tensor_load_to_lds(g0, g1, z4, z4, z8, 0);
#else
    __builtin_amdgcn_tensor_load_to_lds(g0, g1, z4, z4, 0);
#endif
}

// Load 32 consecutive floats (8 x b128) from LDS into a register array.
__device__ static inline void load32(float* dst, const float4* src4) {
#pragma unroll
    for (int c = 0; c < 8; ++c) {
        float4 v = src4[c];
        dst[4 * c + 0] = v.x; dst[4 * c + 1] = v.y;
        dst[4 * c + 2] = v.z; dst[4 * c + 3] = v.w;
    }
}

// ---------------- kernel: one 128-sample tile per block ----------------
__global__ __launch_bounds__(THREADS, 1)
void clifford_geom_product_kernel(const float* __restrict__ x,
                                  const float* __restrict__ y,
                                  float* __restrict__ out,
                                  int n_total) {
    extern __shared__ float lds[];
    const int tid      = (int)threadIdx.x;
    const int tileBase = (int)blockIdx.x * ROWS_PER_BLOCK;
    const int rows     = min(ROWS_PER_BLOCK, n_total - tileBase);

    const uint32_t lds_base = (uint32_t)(uintptr_t)(void*)lds;    // LDS aperture: low 32 bits = LDS offset
    const uint32_t xs_off   = lds_base;
    const uint32_t ys_off   = lds_base + (uint32_t)(ROWS_PER_BLOCK * LDS_STRIDE * sizeof(float));

    // One wave issues both DMA descriptors (TDM ignores EXEC; keep it to a single wave via scalar branch)
    const int waveId = __builtin_amdgcn_readfirstlane(tid) >> 5;
    if (waveId == 0) {
        tdm_load_tile(x + (size_t)tileBase * DIM, xs_off, rows);
        tdm_load_tile(y + (size_t)tileBase * DIM, ys_off, rows);
        __builtin_amdgcn_s_wait_tensorcnt(0);
    }
    __syncthreads();

    const int row = tid;   // lane-per-sample: wave32 -> 32 samples per wave
    const float4* xr4 = (const float4*)(lds + (size_t)row * LDS_STRIDE);
    const float4* yr4 = (const float4*)(lds + (size_t)ROWS_PER_BLOCK * LDS_STRIDE + (size_t)row * LDS_STRIDE);

    const bool live = (tileBase + row) < n_total;
    float* orow = out + ((size_t)tileBase + (size_t)row) * DIM;

    // Zero structure of Cl(3,2,1): sg[k][a]==0 iff bit5(a)=1 && bit5(k)=0.
    // => three dense 32x32 blocks; peak register arrays = 3 x 32 floats (~96 VGPRs).
    float xbuf[32], ybuf[32], acc[32];

    // ---- Phase A: k in [0,32), a in [0,32): x_lo * y_lo ----
    load32(xbuf, xr4);          // x_lo
    load32(ybuf, yr4);          // y_lo
#pragma unroll
    for (int i = 0; i < 32; ++i) acc[i] = 0.0f;
#pragma unroll
    for (int a = 0; a < 32; ++a) {
        const float xa = xbuf[a];
#pragma unroll
        for (int i = 0; i < 32; ++i) {
            const int s = TAB.s[i][a];
            if (s != 0) {
                const float yv = (s > 0) ? ybuf[a ^ i] : -ybuf[a ^ i];
                acc[i] = fmaf(xa, yv, acc[i]);
            }
        }
    }
    if (live) {
#pragma unroll
        for (int c = 0; c < 8; ++c) {
            float4 v;
            v.x = acc[4 * c + 0]; v.y = acc[4 * c + 1];
            v.z = acc[4 * c + 2]; v.w = acc[4 * c + 3];
            ((float4*)orow)[c] = v;
        }
    }

    // ---- Phase B2: k in [32,64), a in [32,64): x_hi * y_lo (a^k stays in low half) ----
    load32(xbuf, xr4 + 8);      // x_hi (ybuf still holds y_lo)
#pragma unroll
    for (int i = 0; i < 32; ++i) acc[i] = 0.0f;
#pragma unroll
    for (int a = 0; a < 32; ++a) {
        const float xa = xbuf[a];
#pragma unroll
        for (int i = 0; i < 32; ++i) {
            const int s = TAB.s[32 + i][32 + a];   // nonzero (bit5(k)=1)
            if (s != 0) {
                const float yv = (s > 0) ? ybuf[a ^ i] : -ybuf[a ^ i];
                acc[i] = fmaf(xa, yv, acc[i]);
            }
        }
    }

    // ---- Phase B1: k in [32,64), a in [0,32): x_lo * y_hi (a^k lands in high half) ----
    load32(xbuf, xr4);          // x_lo again
    load32(ybuf, yr4 + 8);      // y_hi
#pragma unroll
    for (int a = 0; a < 32; ++a) {
        const float xa = xbuf[a];
#pragma unroll
        for (int i = 0; i < 32; ++i) {
            const int s = TAB.s[32 + i][a];        // nonzero (bit5(a)=0)
            if (s != 0) {
                const float yv = (s > 0) ? ybuf[a ^ i] : -ybuf[a ^ i];
                acc[i] = fmaf(xa, yv, acc[i]);
            }
        }
    }
    if (live) {
#pragma unroll
        for (int c = 0; c < 8; ++c) {
            float4 v;
            v.x = acc[4 * c + 0]; v.y = acc[4 * c + 1];
            v.z = acc[4 * c + 2]; v.w = acc[4 * c + 3];
            ((float4*)(orow + 32))[c] = v;
        }
    }
}

extern "C" void kernel_launch(void* const* d_in, const int* in_sizes, int n_in,
                              void* d_out, int out_size, void* d_ws, size_t ws_size,
                              hipStream_t stream) {
    const float* x = (const float*)d_in[0];
    const float* y = (const float*)d_in[1];
    float* out = (float*)d_out;
    const int n_total = in_sizes[0] / DIM;                 // 65536 samples
    const int blocks  = (n_total + ROWS_PER_BLOCK - 1) / ROWS_PER_BLOCK;
    const size_t ldsBytes = 2u * ROWS_PER_BLOCK * LDS_STRIDE * sizeof(float);  // ~68 KB (WGP allows 320 KB)
    clifford_geom_product_kernel<<<blocks, THREADS, ldsBytes, stream>>>(x, y, out, n_total);
}